// FollowingWeightContraction_47004122087962
// MI455X (gfx1250) — compile-verified
//
#include <hip/hip_runtime.h>
#include <hip/hip_bf16.h>

// Problem dims (fixed by the reference):
//   M=1, N=16 (so S = N^3 = 4096 spatial positions), K=23 CG paths,
//   E=10 species, C=128 features, B=256 nodes.
constexpr int KDIM = 23;
constexpr int EDIM = 10;
constexpr int CDIM = 128;
constexpr int BDIM = 256;
constexpr int SDIM = 4096;                 // 16*16*16 (wxvn flattened, w==0)
constexpr long long PER_NODE = (long long)CDIM * SDIM;  // 524288 floats

typedef __attribute__((ext_vector_type(2))) float v2f;
typedef __attribute__((ext_vector_type(4))) float v4f;
typedef __attribute__((ext_vector_type(8))) float v8f;

// ---------------------------------------------------------------------------
// Phase 1: UW[e, c, s] = sum_k W[e, k, c] * U[s, k]      (exact fp32 WMMA)
//   GEMM per e:  A = W[e]^T (C x K),  B = U^T (K x S),  D = (C x S)
//   One wave per 16x16 (c,s) output tile; K=23 padded to 24 -> 6 k-steps of 4.
//   Tiles: E * (C/16) * (S/16) = 10 * 8 * 256 = 20480 waves, 8 waves/block.
// ---------------------------------------------------------------------------
__global__ __launch_bounds__(256)
void FollowingWeightContraction_uw_wmma(const float* __restrict__ U,   // (S, K)
                                        const float* __restrict__ W,   // (E, K, C)
                                        float* __restrict__ UW) {      // (E, C, S)
    const int lane = threadIdx.x & 31;
    const int wid  = threadIdx.x >> 5;
    const int tile = blockIdx.x * 8 + wid;      // < 20480 always

    const int e   = tile >> 11;                 // / (8*256)
    const int rem = tile & 2047;
    const int ct  = rem >> 8;                   // c-tile  [0,8)
    const int st  = rem & 255;                  // s-tile  [0,256)
    const int c0  = ct * 16;
    const int s0  = st * 16;

    const int half = lane >> 4;                 // which lane half (K pair select)
    const int lid  = lane & 15;
    const int cA   = c0 + lid;                  // A-matrix row (M = C dim)
    const int sB   = s0 + lid;                  // B-matrix col (N = S dim)

    v8f acc = {};
#pragma unroll
    for (int kstep = 0; kstep < 6; ++kstep) {
        // 32-bit 16x4 A layout: lanes 0-15 hold K={0,1} in v0,v1; lanes 16-31 K={2,3}.
        // 4x16 B layout mirrors it: lane half selects K pair, vgpr selects K within pair.
        const int kb = kstep * 4 + half * 2;
        v2f a, b;
        a.x = (kb     < KDIM) ? W[(e * KDIM + kb    ) * CDIM + cA] : 0.0f;
        a.y = (kb + 1 < KDIM) ? W[(e * KDIM + kb + 1) * CDIM + cA] : 0.0f;
        b.x = (kb     < KDIM) ? U[(long long)sB * KDIM + kb    ] : 0.0f;
        b.y = (kb + 1 < KDIM) ? U[(long long)sB * KDIM + kb + 1] : 0.0f;
        // D = A x B + C, fp32 exact:  v_wmma_f32_16x16x4_f32
        acc = __builtin_amdgcn_wmma_f32_16x16x4_f32(
            /*neg_a=*/false, a, /*neg_b=*/false, b,
            /*c_mod=*/(short)0, acc, /*reuse_a=*/false, /*reuse_b=*/false);
    }

    // C/D 16x16 layout: vgpr i, lanes 0-15 -> (M=i, N=lane); lanes 16-31 -> (M=i+8).
    float* dst = UW + ((size_t)e * CDIM + (size_t)(c0 + 8 * half)) * SDIM + s0 + lid;
#pragma unroll
    for (int i = 0; i < 8; ++i)
        dst[(size_t)i * SDIM] = acc[i];
}

// ---------------------------------------------------------------------------
// Phase 2: out[b, c, s] = attr[b, e*] * UW[e*, c, s]   (e* = argmax one-hot)
//   512 MB of streaming stores -> non-temporal; UW reads hit L2 (20 MB).
//   float4 per thread; grid = (PER_NODE/4/256, B).
// ---------------------------------------------------------------------------
__global__ __launch_bounds__(256)
void FollowingWeightContraction_bcast(const float* __restrict__ na,  // (B, E)
                                      const float* __restrict__ UW,  // (E, C, S)
                                      float* __restrict__ out) {     // (B, C, S)
    const int b = blockIdx.y;

    // species = argmax over the one-hot row (E=10, L2-cached, cheap)
    float wbest = na[(long long)b * EDIM];
    int e = 0;
#pragma unroll
    for (int i = 1; i < EDIM; ++i) {
        float v = na[(long long)b * EDIM + i];
        if (v > wbest) { wbest = v; e = i; }
    }

    const long long idx = ((long long)blockIdx.x * blockDim.x + threadIdx.x) * 4;
    const v4f* src = (const v4f*)(UW + (long long)e * PER_NODE + idx);
    v4f v = *src;
    v *= wbest;  // one-hot weight (==1.0f), kept for exactness vs reference
    v4f* dst = (v4f*)(out + (long long)b * PER_NODE + idx);
    __builtin_nontemporal_store(v, dst);  // don't pollute L2 holding UW
}

extern "C" void kernel_launch(void* const* d_in, const int* in_sizes, int n_in,
                              void* d_out, int out_size, void* d_ws, size_t ws_size,
                              hipStream_t stream) {
    const float* U  = (const float*)d_in[0];   // (1,16,16,16,23) -> (S,K)
    const float* W  = (const float*)d_in[1];   // (E,K,C)
    const float* na = (const float*)d_in[2];   // (B,E)
    float* out = (float*)d_out;                // (B,C,1,16,16,16) -> (B,C,S)
    float* UW  = (float*)d_ws;                 // (E,C,S) = 20 MB scratch

    // Phase 1: 20480 wave-tiles, 8 waves (256 threads) per block.
    FollowingWeightContraction_uw_wmma<<<20480 / 8, 256, 0, stream>>>(U, W, UW);

    // Phase 2: PER_NODE/4 float4s per node / 256 threads = 512 blocks per node.
    dim3 grid((unsigned)(PER_NODE / 4 / 256), (unsigned)BDIM);
    FollowingWeightContraction_bcast<<<grid, 256, 0, stream>>>(na, UW, out);
}